// Starlet2D_27951647163049
// MI455X (gfx1250) — compile-verified
//
#include <hip/hip_runtime.h>
#include <hip/hip_bf16.h>

#define IMG   1024
#define TSZ   64
#define HALO  30           // 2*(1+2+4+8)
#define RSZ   124          // TSZ + 2*HALO (loaded region)
#define ST    128          // padded LDS row stride (floats) -> 16B-aligned quads
#define OFS   2            // region origin inside padded buffer (makes O's %4==0)
#define NTHREADS 256

// B3-spline taps (exact binary fractions)
#define W0 0.0625f
#define W1 0.25f
#define W2 0.375f

typedef float v4f __attribute__((ext_vector_type(4)));

// ---- CDNA5 async global->LDS path (verified present on this toolchain) ----
#if defined(__has_builtin)
#if __has_builtin(__builtin_amdgcn_global_load_async_to_lds_b32)
#define USE_ASYNC_LDS 1
#endif
#endif

typedef __attribute__((address_space(1))) int global_i32;
typedef __attribute__((address_space(3))) int lds_i32;

__device__ __forceinline__ int refl(int v) {
    v = v < 0 ? -v : v;                     // jnp 'reflect': -1 -> 1
    return v >= IMG ? 2 * IMG - 2 - v : v;  // 1024 -> 1022
}

__device__ __forceinline__ v4f ld4(const float* p) { return *(const v4f*)p; }
__device__ __forceinline__ void st4(float* p, v4f v) { *(v4f*)p = v; }

// One smoothing level: separable 5-tap conv, dilation D, src -> dst.
// Output region [O,O+R)^2 in padded-buffer coords; O % 4 == 0, R % 4 == 0.
// Horizontal covers rows [O-2D, O+R+2D); vertical runs in-place with staging.
template <int D, int O, int R>
__device__ __forceinline__ void starlet_pass(const float* __restrict__ src,
                                             float* __restrict__ dst, int tid) {
    constexpr int C = R / 4;                      // quads per row
    // ---------------- horizontal ----------------
    constexpr int RL = O - 2 * D;
    constexpr int RH = O + R + 2 * D;
    constexpr int HR = RH - RL;
    constexpr int HSTR = NTHREADS / C;            // row strips
    constexpr int HRH = (HR + HSTR - 1) / HSTR;   // rows per strip
    {
        const int qx    = tid % C;
        const int strip = tid / C;
        const int i     = O + 4 * qx;
        if (strip < HSTR) {
            const int j0   = RL + strip * HRH;
            const int jend = (j0 + HRH < RH) ? j0 + HRH : RH;
            const float* p = src + j0 * ST + i;
            float*       q = dst + j0 * ST + i;
            for (int j = j0; j < jend; ++j, p += ST, q += ST) {
                v4f o;
                if constexpr (D >= 4) {
                    // tap vectors are themselves 16B-aligned quads
                    v4f a0 = ld4(p - 2 * D), a1 = ld4(p - D), a2 = ld4(p),
                        a3 = ld4(p + D),     a4 = ld4(p + 2 * D);
                    o = W0 * (a0 + a4) + W1 * (a1 + a3) + W2 * a2;
                } else {
                    // 12-wide aligned register window [x-4, x+8)
                    v4f v0 = ld4(p - 4), v1 = ld4(p), v2 = ld4(p + 4);
                    float w[12];
#pragma unroll
                    for (int m = 0; m < 4; ++m) {
                        w[m] = v0[m]; w[4 + m] = v1[m]; w[8 + m] = v2[m];
                    }
#pragma unroll
                    for (int l = 0; l < 4; ++l)
                        o[l] = W0 * (w[4 + l - 2 * D] + w[4 + l + 2 * D]) +
                               W1 * (w[4 + l - D]     + w[4 + l + D])     +
                               W2 * w[4 + l];
                }
                st4(q, o);
            }
        }
    }
    __syncthreads();
    // ---------------- vertical (in-place, register-staged) ----------------
    constexpr int VSTR = NTHREADS / C;
    constexpr int VRH  = (R + VSTR - 1) / VSTR;
    const int qx    = tid % C;
    const int strip = tid / C;
    const int i     = O + 4 * qx;
    const int j0    = O + strip * VRH;
    const bool act  = (strip < VSTR);
    v4f vals[VRH];
#pragma unroll
    for (int r = 0; r < VRH; ++r) {
        const int j = j0 + r;
        if (act && j < O + R) {
            const float* p = dst + j * ST + i;
            vals[r] = W0 * (ld4(p - 2 * D * ST) + ld4(p + 2 * D * ST)) +
                      W1 * (ld4(p - D * ST)     + ld4(p + D * ST))     +
                      W2 * ld4(p);
        }
    }
    __syncthreads();
#pragma unroll
    for (int r = 0; r < VRH; ++r) {
        const int j = j0 + r;
        if (act && j < O + R)
            st4(dst + j * ST + i, vals[r]);
    }
    __syncthreads();
}

// detail = P - Q over central 64x64 tile (buffer coords [32,96)), float4 stores
__device__ __forceinline__ void write_detail(const float* __restrict__ P,
                                             const float* __restrict__ Q,
                                             float* __restrict__ plane,
                                             int tid, int ox, int oy) {
    const int qx    = tid & 15;        // 16 quads per row
    const int strip = tid >> 4;        // 16 strips x 4 rows
    const int ii    = 4 * qx;
#pragma unroll
    for (int r = 0; r < 4; ++r) {
        const int jj = strip * 4 + r;
        const int li = (32 + jj) * ST + 32 + ii;
        st4(&plane[(size_t)(oy + jj) * IMG + (ox + ii)], ld4(&P[li]) - ld4(&Q[li]));
    }
}

__global__ __launch_bounds__(NTHREADS) void starlet2d_kernel(
    const float* __restrict__ x, float* __restrict__ out) {
    __shared__ __align__(16) float A[ST * ST];
    __shared__ __align__(16) float Bf[ST * ST];

    const int tid = threadIdx.x;
    const int tx = blockIdx.x, ty = blockIdx.y, b = blockIdx.z;
    const int gx0 = tx * TSZ - HALO;
    const int gy0 = ty * TSZ - HALO;
    const float* xb = x + (size_t)b * IMG * IMG;

    // Load 124x124 region (reflect at borders) into A at offset (OFS,OFS).
    // Column fixed per thread (reflect-x computed once); async DMA into LDS.
    {
        const int col   = tid % RSZ;           // 0..123
        const int strip = tid / RSZ;           // 0..1 active (248 threads)
        if (strip < 2) {
            const int gx = refl(gx0 + col);
            const int j0 = strip * (RSZ / 2);  // 62 rows per strip
            for (int r = 0; r < RSZ / 2; ++r) {
                const int j  = j0 + r;
                const int gy = refl(gy0 + j);
                const float* gptr = xb + (size_t)gy * IMG + gx;
                float* lptr = &A[(OFS + j) * ST + OFS + col];
#ifdef USE_ASYNC_LDS
                __builtin_amdgcn_global_load_async_to_lds_b32(
                    (global_i32*)(float*)gptr, (lds_i32*)lptr, 0, 0);
#else
                *lptr = *gptr;
#endif
            }
        }
    }
#ifdef USE_ASYNC_LDS
#if __has_builtin(__builtin_amdgcn_s_wait_asynccnt)
    __builtin_amdgcn_s_wait_asynccnt(0);
#else
    asm volatile("s_wait_asynccnt 0" ::: "memory");
#endif
#endif
    __syncthreads();

    float* outBase = out + (size_t)b * 5 * IMG * IMG;
    const int ox = tx * TSZ, oy = ty * TSZ;
    const size_t PL = (size_t)IMG * IMG;

    // Buffer-coord valid regions: x:[2,126) c1:[4,124) c2:[8,120) c3:[16,112) c4:[32,96)
    starlet_pass<1, 4, 120>(A, Bf, tid);
    write_detail(A, Bf, outBase + 0 * PL, tid, ox, oy);   // d0 = x  - c1
    __syncthreads();

    starlet_pass<2, 8, 112>(Bf, A, tid);
    write_detail(Bf, A, outBase + 1 * PL, tid, ox, oy);   // d1 = c1 - c2
    __syncthreads();

    starlet_pass<4, 16, 96>(A, Bf, tid);
    write_detail(A, Bf, outBase + 2 * PL, tid, ox, oy);   // d2 = c2 - c3
    __syncthreads();

    starlet_pass<8, 32, 64>(Bf, A, tid);
    {   // d3 = c3 - c4 and coarse, fused
        const int qx    = tid & 15;
        const int strip = tid >> 4;
        const int ii    = 4 * qx;
#pragma unroll
        for (int r = 0; r < 4; ++r) {
            const int jj = strip * 4 + r;
            const int li = (32 + jj) * ST + 32 + ii;
            const v4f a  = ld4(&A[li]);
            const size_t go = (size_t)(oy + jj) * IMG + (ox + ii);
            st4(&outBase[3 * PL + go], ld4(&Bf[li]) - a);
            st4(&outBase[4 * PL + go], a);
        }
    }
}

extern "C" void kernel_launch(void* const* d_in, const int* in_sizes, int n_in,
                              void* d_out, int out_size, void* d_ws, size_t ws_size,
                              hipStream_t stream) {
    (void)in_sizes; (void)n_in; (void)d_ws; (void)ws_size; (void)out_size;
    const float* x = (const float*)d_in[0];   // (16,1,1024,1024) f32
    // d_in[1] = kernel_2d (fixed B3 spline, hardcoded), d_in[2] = n_scales (=5)
    float* out = (float*)d_out;               // (16,5,1024,1024) f32

    dim3 grid(IMG / TSZ, IMG / TSZ, 16);      // 16x16 tiles x 16 batches
    dim3 block(NTHREADS);
    starlet2d_kernel<<<grid, block, 0, stream>>>(x, out);
}